// GraphLaplacianAttention_90752658965008
// MI455X (gfx1250) — compile-verified
//
#include <hip/hip_runtime.h>

#define DIM 128
#define LDC 132            // padded row stride (floats) for the row-major sC buffer
#define DEG 16
#define NPW 8              // nodes per group in fused kernel (8 waves, 1 node/wave)
#define SWZ_TILE (128*128) // floats per swizzled 128x128 operand tile (64 KB)
#define PERS_WGS 500       // persistent workgroups (2500 groups -> 5 each)

typedef float v2f __attribute__((ext_vector_type(2)));
typedef float v4f __attribute__((ext_vector_type(4)));
typedef float v8f __attribute__((ext_vector_type(8)));

// ---------------------------------------------------------------------------
// Swizzled operand layouts (fragment-major, 4 contiguous floats per lane per
// K-chunk-pair) so one ds_load_b128 feeds two V_WMMA_F32_16X16X4_F32:
//   A: idx = ((strip*16 + p)*32 + lane)*4 + pos
//   B: idx = ((p*8   + t)*32 + lane)*4 + pos
// with p = k>>3, lane = (m|n & 15) + 16*((k>>1)&1), pos = ((k>>2)&1)*2 + (k&1)
// ---------------------------------------------------------------------------

__device__ __forceinline__ void stage_A_sw(float* s, const float* __restrict__ g,
                                           int tid, int rows) {
  const int row   = tid >> 1;
  const int strip = row >> 4, m = row & 15;
  const int kbase = (tid & 1) * 64;
  const bool ok   = row < rows;
  #pragma unroll
  for (int i = 0; i < 16; ++i) {
    const int k = kbase + i * 4;
    float4 v = make_float4(0.f, 0.f, 0.f, 0.f);
    if (ok) v = *(const float4*)(g + (size_t)row * DIM + k);
    const int p = k >> 3, hi = (k >> 2) & 1;
    float* base = s + (size_t)((strip * 16 + p) * 32) * 4 + hi * 2;
    *(float2*)(base + m * 4)        = make_float2(v.x, v.y);   // k, k+1   -> lane m
    *(float2*)(base + (m + 16) * 4) = make_float2(v.z, v.w);   // k+2, k+3 -> lane m+16
  }
}

__device__ __forceinline__ void stage_B_sw(float* s, const float* __restrict__ g, int tid) {
  const int k     = tid >> 1;
  const int nbase = (tid & 1) * 64;
  const int p   = k >> 3, q = k & 7;
  const int lo  = 16 * ((q >> 1) & 1);
  const int pos = (q >> 2) * 2 + (q & 1);
  #pragma unroll
  for (int i = 0; i < 16; ++i) {
    const int n = nbase + i * 4;
    const float4 v = *(const float4*)(g + (size_t)k * DIM + n);
    const int t = n >> 4, nm = n & 15;
    float* base = s + (size_t)((p * 8 + t) * 32) * 4 + pos;
    base[(nm + lo) * 4]     = v.x;
    base[(nm + 1 + lo) * 4] = v.y;
    base[(nm + 2 + lo) * 4] = v.z;
    base[(nm + 3 + lo) * 4] = v.w;
  }
}

// ---------------------------------------------------------------------------
// One 16-row strip of [128x128]@[128x128] fp32: 256 WMMAs, 144 ds_load_b128.
// Two uniform base pointers + compile-time immediate DS offsets.
// ---------------------------------------------------------------------------
__device__ __forceinline__ void strip_gemm_sw(const float* sA, const float* sB,
                                              int strip, int lane, v8f acc[8]) {
  const float* baseA = sA + (size_t)(strip * 16 * 32 + lane) * 4;
  const float* baseB = sB + (size_t)lane * 4;
  #pragma unroll
  for (int p = 0; p < 16; ++p) {
    const v4f av = *(const v4f*)(baseA + p * 128);
    const v2f a01 = __builtin_shufflevector(av, av, 0, 1);
    const v2f a23 = __builtin_shufflevector(av, av, 2, 3);
    v4f bv[8];
    #pragma unroll
    for (int t = 0; t < 8; ++t)
      bv[t] = *(const v4f*)(baseB + (p * 8 + t) * 128);
    #pragma unroll
    for (int t = 0; t < 8; ++t) {
      const v2f b01 = __builtin_shufflevector(bv[t], bv[t], 0, 1);
      acc[t] = __builtin_amdgcn_wmma_f32_16x16x4_f32(false, a01, false, b01,
                                                     (short)0, acc[t], false, false);
    }
    #pragma unroll
    for (int t = 0; t < 8; ++t) {
      const v2f b23 = __builtin_shufflevector(bv[t], bv[t], 2, 3);
      acc[t] = __builtin_amdgcn_wmma_f32_16x16x4_f32(false, a23, false, b23,
                                                     (short)0, acc[t], false, false);
    }
  }
}

__device__ __forceinline__ void zero_acc(v8f acc[8]) {
  #pragma unroll
  for (int t = 0; t < 8; ++t) { v8f z = {}; acc[t] = z; }
}

// ---------------------------------------------------------------------------
// Node projections: stage x tile once, run 3 GEMMs (Wq, Wk, Wv).
// ---------------------------------------------------------------------------
__global__ void k_nodeproj(const float* __restrict__ x,
                           const float* __restrict__ Wq, const float* __restrict__ Wk,
                           const float* __restrict__ Wv,
                           float* __restrict__ XQ, float* __restrict__ XK,
                           float* __restrict__ XV, int M) {
  extern __shared__ float lds[];
  float* sA = lds;
  float* sB = lds + SWZ_TILE;
  const int tid = threadIdx.x;
  const int m0  = blockIdx.x * 128;
  const int wave = tid >> 5, lane = tid & 31;
  const int lhalf = lane >> 4, lm = lane & 15;

  stage_A_sw(sA, x + (size_t)m0 * DIM, tid, M - m0);

  #pragma unroll
  for (int j = 0; j < 3; ++j) {
    const float* W = (j == 0) ? Wq : (j == 1) ? Wk : Wv;
    float* C       = (j == 0) ? XQ : (j == 1) ? XK : XV;
    __syncthreads();                  // sA staged / previous sB consumers done
    stage_B_sw(sB, W, tid);
    __syncthreads();
    v8f acc[8];
    zero_acc(acc);
    strip_gemm_sw(sA, sB, wave, lane, acc);
    #pragma unroll
    for (int t = 0; t < 8; ++t) {
      const int c = t * 16 + lm;
      #pragma unroll
      for (int i = 0; i < 8; ++i) {
        const int r = m0 + wave * 16 + i + 8 * lhalf;
        if (r < M) C[(size_t)r * DIM + c] = acc[t][i];
      }
    }
  }
}

// ---------------------------------------------------------------------------
// C[M x 128] = A[M x 128] @ B[128 x 128] + bias (final output projection)
// ---------------------------------------------------------------------------
__global__ void k_gemm128(const float* __restrict__ A, const float* __restrict__ B,
                          float* __restrict__ C, int M, const float* __restrict__ bias) {
  extern __shared__ float lds[];
  float* sA = lds;
  float* sB = lds + SWZ_TILE;
  const int tid = threadIdx.x;
  const int m0  = blockIdx.x * 128;

  stage_A_sw(sA, A + (size_t)m0 * DIM, tid, M - m0);
  stage_B_sw(sB, B, tid);
  __syncthreads();

  const int wave = tid >> 5, lane = tid & 31;
  const int lhalf = lane >> 4, lm = lane & 15;
  v8f acc[8];
  zero_acc(acc);
  strip_gemm_sw(sA, sB, wave, lane, acc);

  #pragma unroll
  for (int t = 0; t < 8; ++t) {
    const int c = t * 16 + lm;
    const float bv = bias ? bias[c] : 0.f;
    #pragma unroll
    for (int i = 0; i < 8; ++i) {
      const int r = m0 + wave * 16 + i + 8 * lhalf;
      if (r < M) C[(size_t)r * DIM + c] = acc[t][i] + bv;
    }
  }
}

// ---------------------------------------------------------------------------
// Persistent fused edge pipeline.  Each WG stages Wek+Wev+Wexp into LDS once,
// then loops over 8-node groups (128 contiguous edges each, wave w = node w):
//   EK = edges@Wek (WMMA regs) ; k = SCALE*(EK + XK[dst])  -> sC
//   per-head scores -> @Wexp -> 16-edge segment softmax
//   EV = edges@Wev (WMMA regs) ; v = EV + XV[dst] (sC reused)
//   tmp[n] = sum_e (1[src==dst] - a_e) * v_e
// Next group's edge tile is warmed with global_prefetch during compute.
// ---------------------------------------------------------------------------
__global__ void k_attn_fused(const float* __restrict__ edges,
                             const int*   __restrict__ eidx,   // [2, E]
                             const float* __restrict__ Wek,
                             const float* __restrict__ Wev,
                             const float* __restrict__ Wexp,   // [8,8]
                             const float* __restrict__ XQ,
                             const float* __restrict__ XK,
                             const float* __restrict__ XV,
                             float* __restrict__ tmp,
                             int nGroups, int Etot, float scale) {
  extern __shared__ float lds[];
  float* sEk = lds;                       // swizzled Wek (B operand, resident)
  float* sEv = sEk + SWZ_TILE;            // swizzled Wev (B operand, resident)
  float* sE  = sEv + SWZ_TILE;            // swizzled edge tile (A operand, per group)
  float* sC  = sE + SWZ_TILE;             // 128 x LDC row-major: XK gather -> k -> XV
  float* sQ  = sC + 128 * LDC;            // 8 x 128
  float* sS  = sQ + NPW * DIM;            // 8 x 16 x 8  raw head scores
  float* sP  = sS + NPW * DEG * 8;        // 8 x 16 x 8  expanded scores -> coef
  float* sX  = sP + NPW * DEG * 8;        // 64          Wexp
  float* sM  = sX + 64;                   // 8 x 8       softmax max
  float* sD  = sM + NPW * 8;              // 8 x 8       softmax denom

  const int tid   = threadIdx.x;
  const int wave  = tid >> 5, lane = tid & 31;
  const int lhalf = lane >> 4, lm = lane & 15;

  // ---- resident weights: staged once per WG -------------------------------
  stage_B_sw(sEk, Wek, tid);
  stage_B_sw(sEv, Wev, tid);
  if (tid < 64) sX[tid] = Wexp[tid];

  for (int g = blockIdx.x; g < nGroups; g += gridDim.x) {
    const int n0 = g * NPW;
    const int e0 = n0 * DEG;

    __syncthreads();                     // prev group done with sE/sC/sQ/sS/sP

    // ---- stage per-group inputs -------------------------------------------
    stage_A_sw(sE, edges + (size_t)e0 * DIM, tid, 128);
    ((float4*)sQ)[tid] = ((const float4*)(XQ + (size_t)n0 * DIM))[tid];
    {   // gather XK[dst] for all 128 edges -> sC (row-major)
      const int row = tid >> 1, half = (tid & 1) * 64;
      const int dst = eidx[Etot + e0 + row];
      const float4* gp = (const float4*)(XK + (size_t)dst * DIM + half);
      float4* sp = (float4*)(sC + row * LDC + half);
      #pragma unroll
      for (int i = 0; i < 16; ++i) sp[i] = gp[i];
    }
    __syncthreads();

    // ---- warm next group's edge tile (fire-and-forget, counter-free) -----
    {
      const int gn = g + gridDim.x;
      if (gn < nGroups) {
        const float* nxt = edges + (size_t)gn * NPW * DEG * DIM;
        __builtin_prefetch(nxt + (size_t)tid * 64, 0, 3);       // 256 B/thread
        __builtin_prefetch(nxt + (size_t)tid * 64 + 32, 0, 3);  // (2 cachelines)
      }
    }

    // ---- GEMM1: EK ; k = scale*(EK + XK[dst]) into sC (own rows) ---------
    v8f acc[8];
    zero_acc(acc);
    strip_gemm_sw(sE, sEk, wave, lane, acc);
    #pragma unroll
    for (int t = 0; t < 8; ++t) {
      #pragma unroll
      for (int i = 0; i < 8; ++i) {
        const int le = wave * DEG + i + 8 * lhalf;
        const int c  = t * 16 + lm;
        sC[le * LDC + c] = scale * (acc[t][i] + sC[le * LDC + c]);
      }
    }

    // ---- per-head scores (wave w handles node w) --------------------------
    #pragma unroll
    for (int rep = 0; rep < 4; ++rep) {
      const int idx = rep * 32 + lane;   // 128 (e,h) dots per node
      const int e = idx >> 3, h = idx & 7;
      const float* qp = sQ + wave * DIM + h * 16;
      const float* kp = sC + (wave * DEG + e) * LDC + h * 16;
      float d = 0.f;
      #pragma unroll
      for (int j = 0; j < 16; ++j) d += qp[j] * kp[j];
      sS[(wave * DEG + e) * 8 + h] = d;
    }
    __syncthreads();

    // ---- head expansion: s2 = s @ Wexp ------------------------------------
    #pragma unroll
    for (int rep = 0; rep < 4; ++rep) {
      const int idx = rep * 32 + lane;
      const int e = idx >> 3, he = idx & 7;
      float v = 0.f;
      #pragma unroll
      for (int h = 0; h < 8; ++h) v += sS[(wave * DEG + e) * 8 + h] * sX[h * 8 + he];
      sP[(wave * DEG + e) * 8 + he] = v;
    }
    __syncthreads();

    // ---- segment softmax stats over 16 edges of each (node, he) -----------
    if (tid < NPW * 8) {
      const int w2 = tid >> 3, he = tid & 7;
      float m = -3.402823466e38f;
      for (int e = 0; e < DEG; ++e) m = fmaxf(m, sP[(w2 * DEG + e) * 8 + he]);
      float s = 0.f;
      for (int e = 0; e < DEG; ++e) s += __expf(sP[(w2 * DEG + e) * 8 + he] - m);
      sM[tid] = m;
      sD[tid] = s;
    }
    __syncthreads();

    // ---- coefficients: coef = 1[src==dst] - a -----------------------------
    #pragma unroll
    for (int rep = 0; rep < 4; ++rep) {
      const int idx = rep * 32 + lane;
      const int e = idx >> 3, he = idx & 7;
      const int ge = e0 + wave * DEG + e;
      const float a = __expf(sP[(wave * DEG + e) * 8 + he] - sM[wave * 8 + he])
                      / sD[wave * 8 + he];
      const float selfc = (eidx[ge] == eidx[Etot + ge]) ? 1.f : 0.f;
      sP[(wave * DEG + e) * 8 + he] = selfc - a;
    }
    {   // gather XV[dst] into sC (own rows; k-values are dead now)
      const int row = wave * DEG + (lane >> 1), half = (lane & 1) * 64;
      const int dst = eidx[Etot + e0 + row];
      const float4* gp = (const float4*)(XV + (size_t)dst * DIM + half);
      float4* sp = (float4*)(sC + row * LDC + half);
      #pragma unroll
      for (int i = 0; i < 16; ++i) sp[i] = gp[i];
    }
    __syncthreads();                     // coefs & XV visible across lanes

    // ---- GEMM2: EV --------------------------------------------------------
    zero_acc(acc);
    strip_gemm_sw(sE, sEv, wave, lane, acc);

    // ---- contraction: tmp[n,dim] = sum_e coef_e * (XV+EV)[e,dim] ----------
    #pragma unroll
    for (int t = 0; t < 8; ++t) {        // t == expanded-head index
      float sum = 0.f;
      #pragma unroll
      for (int i = 0; i < 8; ++i) {
        const int e = i + 8 * lhalf;     // acc row -> local edge
        const float coef = sP[(wave * DEG + e) * 8 + t];
        const float v = sC[(wave * DEG + e) * LDC + t * 16 + lm] + acc[t][i];
        sum += coef * v;
      }
      sum += __shfl_xor(sum, 16);        // combine complementary 8-edge halves
      if (lhalf == 0) tmp[(size_t)(n0 + wave) * DIM + t * 16 + lm] = sum;
    }
  }
}

// ---------------------------------------------------------------------------
extern "C" void kernel_launch(void* const* d_in, const int* in_sizes, int n_in,
                              void* d_out, int out_size, void* d_ws, size_t ws_size,
                              hipStream_t stream) {
  (void)n_in; (void)out_size; (void)ws_size;
  const float* x     = (const float*)d_in[0];
  const float* edges = (const float*)d_in[1];
  const int*   eidx  = (const int*)  d_in[2];
  const float* Wq    = (const float*)d_in[3];
  const float* Wk    = (const float*)d_in[4];
  const float* Wv    = (const float*)d_in[5];
  const float* Wek   = (const float*)d_in[6];
  const float* Wev   = (const float*)d_in[7];
  const float* Wexp  = (const float*)d_in[8];
  const float* Wout  = (const float*)d_in[9];
  const float* bout  = (const float*)d_in[10];

  const int N = in_sizes[0] / DIM;       // 20000
  const int E = in_sizes[1] / DIM;       // 320000
  const float scale = 0.25f;             // HD^-0.5, HD = 16

  const size_t nd = (size_t)N * DIM;
  float* XQ  = (float*)d_ws;
  float* XK  = XQ + nd;
  float* XV  = XK + nd;
  float* tmp = XV + nd;

  const size_t shG = (size_t)2 * SWZ_TILE * sizeof(float);                  // 128 KB
  const size_t shF = (size_t)(3 * SWZ_TILE + 128 * LDC + NPW * DIM
                              + 2 * NPW * DEG * 8 + 64 + 2 * NPW * 8)
                     * sizeof(float);                                       // ~277 KB
  hipFuncSetAttribute(reinterpret_cast<const void*>(k_nodeproj),
                      hipFuncAttributeMaxDynamicSharedMemorySize, (int)shG);
  hipFuncSetAttribute(reinterpret_cast<const void*>(k_gemm128),
                      hipFuncAttributeMaxDynamicSharedMemorySize, (int)shG);
  hipFuncSetAttribute(reinterpret_cast<const void*>(k_attn_fused),
                      hipFuncAttributeMaxDynamicSharedMemorySize, (int)shF);

  const int gBlocks = (N + 127) / 128;
  const int nGroups = N / NPW;           // 2500
  const int persWGs = (nGroups < PERS_WGS) ? nGroups : PERS_WGS;
  k_nodeproj<<<gBlocks, 256, shG, stream>>>(x, Wq, Wk, Wv, XQ, XK, XV, N);
  k_attn_fused<<<persWGs, 256, shF, stream>>>(edges, eidx, Wek, Wev, Wexp,
                                              XQ, XK, XV, tmp, nGroups, E, scale);
  k_gemm128<<<gBlocks, 256, shG, stream>>>(tmp, Wout, (float*)d_out, N, bout);
}